// Fusion_Module_33371895890377
// MI455X (gfx1250) — compile-verified
//
#include <hip/hip_runtime.h>
#include <math.h>
#include <stdint.h>

// ============================================================================
// MI455X (gfx1250) implementation of the 4-scale cross-modal SS2D fusion block.
// - All dense contractions on V_WMMA_F32_16X16X4_F32 (native f32 WMMA; exact
//   precision vs. the f32 reference). 16x64 tile per wave, branchless clamped
//   fragment loads (invalid N columns are computed but never stored).
// - Selective scan: per-channel recurrence in VGPRs; Bs/Cs tiles staged into
//   LDS via GLOBAL_LOAD_ASYNC_TO_LDS_B128 + S_WAIT_ASYNCCNT (CDNA5 async path).
// - Whole working set (<310MB) is L2/HBM friendly at 23.3 TB/s; total math is
//   only ~20 GF, so the design optimizes launch-count, fragment reuse and the
//   sequential-scan latency path rather than peak TFLOPs.
// ============================================================================

typedef float v2f __attribute__((ext_vector_type(2)));
typedef float v8f __attribute__((ext_vector_type(8)));

// ---------------------------------------------------------------------------
// Generic WMMA f32 GEMM:  D[M,N] = epilogue( A[M,K] * W[N,K]^T )
//   AMODE 0: A row-major with leading dim lda
//   AMODE 1: A is NCHW read as [M=(b,hw), K=c]: addr=(m/a_hw)*a_bs+k*a_hw+m%a_hw
//   dmode 0: D row-major, ldd = N
//   dmode 1: D NCHW: addr=(m/d_hw)*d_bs+(n+d_coff)*d_hw+m%d_hw
//   epi 0: store; 1: relu((x+bias[n])*(g[n]/sqrt(1+1e-5))+bb[n]); 2: x+res[NCHW]
// One wave per block; 16x64 output tile = 4 WMMA tiles sharing one A fragment.
// Fragment layout per ISA 7.12.2: lane q = row/col, lane-half selects K pair,
// so both A (row-major) and W fragments are contiguous 8B loads.
// ---------------------------------------------------------------------------
__device__ __forceinline__ void gemm_epilogue(
    v8f acc, int nc, int m0, int half, int N,
    float* __restrict__ D,
    const float* __restrict__ bias, const float* __restrict__ bng,
    const float* __restrict__ bnb, const float* __restrict__ res,
    int dmode, int d_bs, int d_hw, int d_coff, int r_bs, int epi)
{
  if (nc >= N) return;
  const float BNS = 0.99999500003749969f; // 1/sqrt(1+1e-5)
#pragma unroll
  for (int r = 0; r < 8; ++r) {
    int row = m0 + 8 * half + r;
    float x = acc[r];
    long daddr;
    if (dmode == 0) daddr = (long)row * N + nc;
    else            daddr = (long)(row / d_hw) * d_bs + (long)(nc + d_coff) * d_hw + (row % d_hw);
    if (epi == 1) {
      x = (x + bias[nc]) * (bng[nc] * BNS) + bnb[nc];
      x = fmaxf(x, 0.f);
    } else if (epi == 2) {
      x += res[(long)(row / d_hw) * r_bs + (long)nc * d_hw + (row % d_hw)];
    }
    D[daddr] = x;
  }
}

template <int AMODE>
__global__ void __launch_bounds__(32) wmma_gemm_f32(
    const float* __restrict__ A, const float* __restrict__ W,
    float* __restrict__ D,
    const float* __restrict__ bias, const float* __restrict__ bng,
    const float* __restrict__ bnb, const float* __restrict__ res,
    int M, int N, int K, int lda,
    int a_bs, int a_hw,
    int dmode, int d_bs, int d_hw, int d_coff, int r_bs,
    int epi)
{
  const int m0 = blockIdx.x * 16;
  const int n0 = blockIdx.y * 64;
  const int t = threadIdx.x;
  const int half = t >> 4;    // selects K pair (k+2*half, k+2*half+1)
  const int q = t & 15;       // row (A) / column (W) within a tile

  v8f acc0 = {0.f,0.f,0.f,0.f,0.f,0.f,0.f,0.f};
  v8f acc1 = acc0, acc2 = acc0, acc3 = acc0;

  const int arow = m0 + q;    // M is a multiple of 16 -> always valid
  long abase;
  if (AMODE == 0) abase = (long)arow * lda;
  else            abase = (long)(arow / a_hw) * a_bs + (arow % a_hw);
  const float* ap = A + abase;

  // Clamp invalid weight rows to row N-1: out-of-range D columns receive
  // garbage but are discarded by the epilogue guard -> branchless loads.
  const int nc0 = n0 + q,      nc1 = n0 + 16 + q;
  const int nc2 = n0 + 32 + q, nc3 = n0 + 48 + q;
  const float* w0 = W + (long)(nc0 < N ? nc0 : N - 1) * K;
  const float* w1 = W + (long)(nc1 < N ? nc1 : N - 1) * K;
  const float* w2 = W + (long)(nc2 < N ? nc2 : N - 1) * K;
  const float* w3 = W + (long)(nc3 < N ? nc3 : N - 1) * K;

#pragma unroll 2
  for (int k = 0; k < K; k += 4) {   // K is always a multiple of 4 here
    const int ka = k + 2 * half;
    v2f a;
    if (AMODE == 0) {
      a = *(const v2f*)(ap + ka);                  // 8B-aligned contiguous
    } else {
      a.x = ap[(long)ka * a_hw];                   // NCHW implicit transpose
      a.y = ap[(long)(ka + 1) * a_hw];
    }
    v2f b0 = *(const v2f*)(w0 + ka);
    v2f b1 = *(const v2f*)(w1 + ka);
    v2f b2 = *(const v2f*)(w2 + ka);
    v2f b3 = *(const v2f*)(w3 + ka);
    // EXEC is all-ones (uniform control flow); A fragment reused 4x.
    acc0 = __builtin_amdgcn_wmma_f32_16x16x4_f32(false, a, false, b0, (short)0, acc0, false, false);
    acc1 = __builtin_amdgcn_wmma_f32_16x16x4_f32(false, a, false, b1, (short)0, acc1, false, false);
    acc2 = __builtin_amdgcn_wmma_f32_16x16x4_f32(false, a, false, b2, (short)0, acc2, false, false);
    acc3 = __builtin_amdgcn_wmma_f32_16x16x4_f32(false, a, false, b3, (short)0, acc3, false, false);
  }

  gemm_epilogue(acc0, nc0, m0, half, N, D, bias, bng, bnb, res, dmode, d_bs, d_hw, d_coff, r_bs, epi);
  gemm_epilogue(acc1, nc1, m0, half, N, D, bias, bng, bnb, res, dmode, d_bs, d_hw, d_coff, r_bs, epi);
  gemm_epilogue(acc2, nc2, m0, half, N, D, bias, bng, bnb, res, dmode, d_bs, d_hw, d_coff, r_bs, epi);
  gemm_epilogue(acc3, nc3, m0, half, N, D, bias, bng, bnb, res, dmode, d_bs, d_hw, d_coff, r_bs, epi);
}

// ---------------------------------------------------------------------------
// im2row for c1 (3x3, pad 1) over concat([rgb,t], channel): A1[M=(b,hw)][K=2C*9]
// ---------------------------------------------------------------------------
__global__ void __launch_bounds__(256) im2row_kernel(
    const float* __restrict__ rgb, const float* __restrict__ tin,
    float* __restrict__ A1, int Bn, int S, int C)
{
  const int K = 2 * C * 9, L = S * S;
  long idx = (long)blockIdx.x * blockDim.x + threadIdx.x;
  long total = (long)Bn * L * K;
  if (idx >= total) return;
  int col = (int)(idx % K);
  long m = idx / K;
  int hw = (int)(m % L);
  int b = (int)(m / L);
  int c = col / 9, j = col % 9;
  int h = hw / S + j / 3 - 1;
  int w = hw % S + j % 3 - 1;
  float v = 0.f;
  if (h >= 0 && h < S && w >= 0 && w < S) {
    const float* src = (c < C) ? rgb : tin;
    int cc = (c < C) ? c : c - C;
    v = src[((long)b * C + cc) * L + (long)h * S + w];
  }
  A1[idx] = v;
}

// ---------------------------------------------------------------------------
// Depthwise 3x3 (SAME) + bias + SiLU on columns [0,Din) of a [B*L, 2*Din] map.
// ---------------------------------------------------------------------------
__global__ void __launch_bounds__(256) dwconv_silu_kernel(
    const float* __restrict__ xp, const float* __restrict__ cw,
    const float* __restrict__ cb, float* __restrict__ out,
    int Bn, int S, int Din)
{
  long idx = (long)blockIdx.x * blockDim.x + threadIdx.x;
  long total = (long)Bn * S * S * Din;
  if (idx >= total) return;
  int c = (int)(idx % Din);
  long m = idx / Din;
  int hw = (int)(m % (S * S));
  int b = (int)(m / (S * S));
  int h = hw / S, w = hw % S;
  const int stride = 2 * Din;
  float acc = cb[c];
#pragma unroll
  for (int dy = 0; dy < 3; ++dy)
#pragma unroll
    for (int dx = 0; dx < 3; ++dx) {
      int hh = h + dy - 1, ww = w + dx - 1;
      if (hh >= 0 && hh < S && ww >= 0 && ww < S)
        acc += xp[((long)b * S * S + (long)hh * S + ww) * stride + c] * cw[c * 9 + dy * 3 + dx];
    }
  acc = acc / (1.f + __expf(-acc));  // SiLU
  out[idx] = acc;
}

// ---------------------------------------------------------------------------
// Build xs[k][b][l2][d] for the 4 scan directions.
// k=0: (h,w) row-major; k=1: (w,h); l2 interleaves rgb(0)/t(1); k+2 = flipped.
// ---------------------------------------------------------------------------
__global__ void __launch_bounds__(256) build_xs_kernel(
    const float* __restrict__ r1, const float* __restrict__ t1,
    float* __restrict__ xs, int Bn, int S, int Din)
{
  const int L2 = 2 * S * S;
  long idx = (long)blockIdx.x * blockDim.x + threadIdx.x;
  long total = 2L * Bn * L2 * Din;
  if (idx >= total) return;
  int d = (int)(idx % Din);
  long r = idx / Din;
  int l2 = (int)(r % L2); r /= L2;
  int b = (int)(r % Bn);
  int k = (int)(r / Bn);            // 0 or 1
  int l = l2 >> 1, mm = l2 & 1;
  int h, w;
  if (k == 0) { h = l / S; w = l % S; } else { w = l / S; h = l % S; }
  const float* src = mm ? t1 : r1;
  float v = src[((long)b * S * S + (long)h * S + w) * Din + d];
  long lane = (long)l2 * Din + d;
  long lane_f = (long)(L2 - 1 - l2) * Din + d;
  xs[((long)(k * Bn + b)) * L2 * Din + lane] = v;
  xs[((long)((k + 2) * Bn + b)) * L2 * Din + lane_f] = v;
}

// ---------------------------------------------------------------------------
// Selective scan. One block per (k,b); each thread owns one channel d and its
// 16-entry state. A = -exp(A_logs). Bs/Cs (shared by all d in a (k,b)) are
// staged into LDS with the CDNA5 async DMA path:
//   global_load_async_to_lds_b128  +  s_wait_asynccnt 0  +  barrier
// All chunk addresses are 16B aligned (R and R2N are multiples of 4 floats).
// ---------------------------------------------------------------------------
#define SCAN_TILE 64
__global__ void __launch_bounds__(256) scan_kernel(
    const float* __restrict__ xs, const float* __restrict__ dts,
    const float* __restrict__ xdbl, const float* __restrict__ Alogs,
    const float* __restrict__ Ds, const float* __restrict__ dtb,
    float* __restrict__ ys, int Bn, int Din, int L2, int R, int R2N)
{
  __shared__ __align__(16) float sB[SCAN_TILE][16];
  __shared__ __align__(16) float sC[SCAN_TILE][16];
  const int kb = blockIdx.x;
  const int k = kb / Bn, b = kb % Bn;
  const float* xsk = xs + ((long)(k * Bn + b)) * L2 * Din;
  const float* dtk = dts + ((long)(k * Bn + b)) * L2 * Din;
  const float* xd = xdbl + ((long)(k * Bn + b)) * L2 * R2N;
  float* ysk = ys + ((long)(k * Bn + b)) * L2 * Din;

  const unsigned ldsB = (unsigned)(uintptr_t)&sB[0][0];
  const unsigned ldsC = (unsigned)(uintptr_t)&sC[0][0];

  for (int d0 = 0; d0 < Din; d0 += 256) {
    const int d = d0 + threadIdx.x;
    const bool act = d < Din;
    float An[16], h[16];
    float Dv = 0.f, db = 0.f;
    if (act) {
      Dv = Ds[k * Din + d];
      db = dtb[k * Din + d];
#pragma unroll
      for (int n = 0; n < 16; ++n) An[n] = -__expf(Alogs[(long)(k * Din + d) * 16 + n]);
    }
#pragma unroll
    for (int n = 0; n < 16; ++n) h[n] = 0.f;

    for (int l0 = 0; l0 < L2; l0 += SCAN_TILE) {
      __syncthreads();   // previous tile fully consumed before DMA overwrite
      // 64 rows x 32 floats (Bs|Cs) = 512 x 16B chunks; 2 per thread.
      for (int ch = threadIdx.x; ch < SCAN_TILE * 8; ch += 256) {
        int rr = ch >> 3;
        int cq = (ch & 7) * 4;  // float offset within the 32-float row slice
        unsigned long long src =
            (unsigned long long)(uintptr_t)(xd + (long)(l0 + rr) * R2N + R + cq);
        unsigned dst = (cq < 16) ? (ldsB + ((unsigned)(rr * 16 + cq) << 2))
                                 : (ldsC + ((unsigned)(rr * 16 + cq - 16) << 2));
        asm volatile("global_load_async_to_lds_b128 %0, %1, off"
                     :: "v"(dst), "v"(src) : "memory");
      }
      asm volatile("s_wait_asynccnt 0x0" ::: "memory");  // own DMAs landed
      __syncthreads();                                   // everyone's landed
      if (act) {
        for (int li = 0; li < SCAN_TILE; ++li) {
          long off = (long)(l0 + li) * Din + d;
          float u = xsk[off];
          float dtv = dtk[off] + db;
          float delta = (dtv > 20.f) ? dtv : log1pf(__expf(dtv)); // softplus
          float du = delta * u;
          float acc = 0.f;
#pragma unroll
          for (int n = 0; n < 16; ++n) {
            float dA = __expf(delta * An[n]);
            h[n] = h[n] * dA + du * sB[li][n];
            acc += h[n] * sC[li][n];
          }
          ysk[off] = acc + Dv * u;
        }
      }
    }
  }
}

// ---------------------------------------------------------------------------
// Merge the 4 directional outputs (flip + whmap index mapping of the
// reference) into the rgb (even channels) / t (odd channels) branch feature
// maps fr/ft, each [B*L, Din], replicating the reference's exact
// (Din/2, L2) -> (H, W, Din) reinterpretation.
// ---------------------------------------------------------------------------
__global__ void __launch_bounds__(256) combine_kernel(
    const float* __restrict__ ys, float* __restrict__ fr, float* __restrict__ ft,
    int Bn, int S, int Din)
{
  const int L = S * S, L2 = 2 * L, HW = L;
  const long per_b = (long)Din * L;   // == (Din/2)*L2
  long idx = (long)blockIdx.x * blockDim.x + threadIdx.x;
  long total = (long)Bn * per_b;
  if (idx >= total) return;
  long i = idx % per_b;
  int b = (int)(idx / per_b);
  long c2 = i / L2;
  int pos = (int)(i % L2);
  const long kstride = (long)L2 * Din;
  const float* y0 = ys + ((long)(0 * Bn + b)) * kstride;
  const float* y1 = ys + ((long)(1 * Bn + b)) * kstride;
  const float* y2 = ys + ((long)(2 * Bn + b)) * kstride;
  const float* y3 = ys + ((long)(3 * Bn + b)) * kstride;

#pragma unroll
  for (int mm = 0; mm < 2; ++mm) {
    int d = (int)(2 * c2) + mm;
    float v = y0[(long)pos * Din + d] + y2[(long)(L2 - 1 - pos) * Din + d];
    // whmap: out[c,h,w] = in[c,w,h] on the (2*Din, W, H) view of (Din, L2)
    long j = (long)d * L2 + pos;
    long c = j / HW;
    int rem = (int)(j % HW);
    int h = rem / S, w = rem % S;
    long jin = c * HW + (long)w * S + h;
    int ind = (int)(jin / L2);
    int inl = (int)(jin % L2);
    v += y1[(long)inl * Din + ind];
    v += y3[(long)(L2 - 1 - inl) * Din + ind];
    (mm ? ft : fr)[(long)b * per_b + i] = v;
  }
}

// ---------------------------------------------------------------------------
// Row LayerNorm over Din + SiLU gate (gate = columns [Din,2Din) of the input
// projection map). In-place on f. One block (256 threads) per row.
// ---------------------------------------------------------------------------
__global__ void __launch_bounds__(256) ln_gate_kernel(
    float* __restrict__ f, const float* __restrict__ gatebuf,
    const float* __restrict__ gam, const float* __restrict__ bet, int Din)
{
  __shared__ float buf[256];
  const long row = blockIdx.x;
  float* fr = f + row * Din;
  const float* gr = gatebuf + row * (2L * Din) + Din;
  const int tid = threadIdx.x;

  float s = 0.f, s2 = 0.f;
  for (int i = tid; i < Din; i += 256) { float v = fr[i]; s += v; s2 += v * v; }
  buf[tid] = s; __syncthreads();
  for (int o = 128; o > 0; o >>= 1) { if (tid < o) buf[tid] += buf[tid + o]; __syncthreads(); }
  float mean = buf[0] / Din; __syncthreads();
  buf[tid] = s2; __syncthreads();
  for (int o = 128; o > 0; o >>= 1) { if (tid < o) buf[tid] += buf[tid + o]; __syncthreads(); }
  float var = buf[0] / Din - mean * mean;
  float rstd = rsqrtf(var + 1e-5f);
  for (int i = tid; i < Din; i += 256) {
    float v = (fr[i] - mean) * rstd * gam[i] + bet[i];
    float g8 = gr[i];
    fr[i] = v * (g8 / (1.f + __expf(-g8)));
  }
}

// ---------------------------------------------------------------------------
// Host orchestration
// ---------------------------------------------------------------------------
static void launch_gemm(hipStream_t st, const float* A, const float* W, float* D,
                        const float* bias, const float* bng, const float* bnb,
                        const float* res,
                        int M, int N, int K, int lda, int amode, int a_bs, int a_hw,
                        int dmode, int d_bs, int d_hw, int d_coff, int r_bs, int epi)
{
  dim3 grid(M / 16, (N + 63) / 64, 1);
  if (amode == 0)
    wmma_gemm_f32<0><<<grid, 32, 0, st>>>(A, W, D, bias, bng, bnb, res, M, N, K, lda,
                                          a_bs, a_hw, dmode, d_bs, d_hw, d_coff, r_bs, epi);
  else
    wmma_gemm_f32<1><<<grid, 32, 0, st>>>(A, W, D, bias, bng, bnb, res, M, N, K, lda,
                                          a_bs, a_hw, dmode, d_bs, d_hw, d_coff, r_bs, epi);
}

extern "C" void kernel_launch(void* const* d_in, const int* in_sizes, int n_in,
                              void* d_out, int out_size, void* d_ws, size_t ws_size,
                              hipStream_t stream)
{
  (void)in_sizes; (void)n_in; (void)out_size; (void)ws_size;
  const int CHS[4] = {64, 128, 256, 512};
  const int SSZ[4] = {64, 32, 16, 8};
  const int Bn = 4;
  float* ws = (float*)d_ws;
  float* out = (float*)d_out;

  // Workspace layout (floats). Sized at scale-0 maxima; ~300 MB total.
  // The im2row arena for the c1 branch aliases the xs+x_dbl region (left
  // branch runs first each scale).
  const long OFF_XS   = 0;                       // 16,777,216
  const long OFF_XDBL = 16777216;                //  4,718,592
  const long OFF_RP   = 21495808;                //  4,194,304
  const long OFF_TP   = OFF_RP + 4194304;
  const long OFF_R1   = OFF_TP + 4194304;        //  2,097,152
  const long OFF_T1   = OFF_R1 + 2097152;
  const long OFF_DTS  = OFF_T1 + 2097152;        // 16,777,216
  const long OFF_YS   = OFF_DTS + 16777216;      // 16,777,216
  const long OFF_FR   = OFF_YS + 16777216;       //  2,097,152
  const long OFF_FT   = OFF_FR + 2097152;
  const long OFF_CAT3 = OFF_FT + 2097152;        //  3,145,728

  long out_off = 0;
  for (int s = 0; s < 4; ++s) {
    const int C = CHS[s], S = SSZ[s];
    const int HW = S * S, L = HW, L2 = 2 * L;
    const int Din = 2 * C, R = C / 16, R2N = R + 32;
    const int M = Bn * L, M2 = Bn * L2;

    const float* rgb = (const float*)d_in[2 * s];
    const float* tin = (const float*)d_in[2 * s + 1];
    const float* P[23];
    for (int j = 0; j < 23; ++j) P[j] = (const float*)d_in[8 + s * 23 + j];
    const float *in1_w = P[0], *in2_w = P[1], *conv_w = P[2], *conv_b = P[3];
    const float *x_proj_w = P[4], *dt_w = P[5], *dt_b = P[6];
    const float *A_logs = P[7], *Ds = P[8];
    const float *ln1_g = P[9], *ln1_b = P[10], *ln2_g = P[11], *ln2_b = P[12];
    const float *out1_w = P[13], *out2_w = P[14];
    const float *c1_w = P[15], *c1_b = P[16], *bn1_g = P[17], *bn1_b = P[18];
    const float *c2_w = P[19], *c2_b = P[20], *bn2_g = P[21], *bn2_b = P[22];

    // ---- left branch: im2row + c1 3x3 conv (+BN+ReLU) -> cat3[:, 0:C] ----
    {
      const int K1 = Din * 9;
      long tot = (long)M * K1;
      im2row_kernel<<<(int)((tot + 255) / 256), 256, 0, stream>>>(rgb, tin, ws + OFF_XS, Bn, S, C);
      launch_gemm(stream, ws + OFF_XS, c1_w, ws + OFF_CAT3, c1_b, bn1_g, bn1_b, nullptr,
                  M, C, K1, K1, 0, 0, 0, 1, 3 * C * HW, HW, 0, 0, 1);
    }

    // ---- input projections (NCHW implicit transpose), [B*L, 4C] ----
    launch_gemm(stream, rgb, in1_w, ws + OFF_RP, nullptr, nullptr, nullptr, nullptr,
                M, 4 * C, C, 0, 1, C * HW, HW, 0, 0, 0, 0, 0, 0);
    launch_gemm(stream, tin, in2_w, ws + OFF_TP, nullptr, nullptr, nullptr, nullptr,
                M, 4 * C, C, 0, 1, C * HW, HW, 0, 0, 0, 0, 0, 0);

    // ---- depthwise 3x3 + SiLU on the first Din columns ----
    {
      long tot = (long)M * Din;
      dwconv_silu_kernel<<<(int)((tot + 255) / 256), 256, 0, stream>>>(
          ws + OFF_RP, conv_w, conv_b, ws + OFF_R1, Bn, S, Din);
      dwconv_silu_kernel<<<(int)((tot + 255) / 256), 256, 0, stream>>>(
          ws + OFF_TP, conv_w, conv_b, ws + OFF_T1, Bn, S, Din);
    }

    // ---- pack 4 scan directions ----
    {
      long tot = 2L * Bn * L2 * Din;
      build_xs_kernel<<<(int)((tot + 255) / 256), 256, 0, stream>>>(
          ws + OFF_R1, ws + OFF_T1, ws + OFF_XS, Bn, S, Din);
    }

    // ---- x_proj and dt GEMMs per direction ----
    for (int k = 0; k < 4; ++k) {
      const float* xsk = ws + OFF_XS + (long)k * Bn * L2 * Din;
      float* xdk = ws + OFF_XDBL + (long)k * Bn * L2 * R2N;
      launch_gemm(stream, xsk, x_proj_w + (long)k * R2N * Din, xdk,
                  nullptr, nullptr, nullptr, nullptr,
                  M2, R2N, Din, Din, 0, 0, 0, 0, 0, 0, 0, 0, 0);
      launch_gemm(stream, xdk, dt_w + (long)k * Din * R,
                  ws + OFF_DTS + (long)k * Bn * L2 * Din,
                  nullptr, nullptr, nullptr, nullptr,
                  M2, Din, R, R2N, 0, 0, 0, 0, 0, 0, 0, 0, 0);
    }

    // ---- selective scan (async-LDS staged Bs/Cs) ----
    scan_kernel<<<4 * Bn, 256, 0, stream>>>(
        ws + OFF_XS, ws + OFF_DTS, ws + OFF_XDBL, A_logs, Ds, dt_b,
        ws + OFF_YS, Bn, Din, L2, R, R2N);

    // ---- merge directions into fr/ft ----
    {
      long tot = (long)Bn * Din * L;
      combine_kernel<<<(int)((tot + 255) / 256), 256, 0, stream>>>(
          ws + OFF_YS, ws + OFF_FR, ws + OFF_FT, Bn, S, Din);
    }

    // ---- LayerNorm + SiLU gate (in place) ----
    ln_gate_kernel<<<M, 256, 0, stream>>>(ws + OFF_FR, ws + OFF_RP, ln1_g, ln1_b, Din);
    ln_gate_kernel<<<M, 256, 0, stream>>>(ws + OFF_FT, ws + OFF_TP, ln2_g, ln2_b, Din);

    // ---- output projections + residual -> cat3[:, C:2C] and [:, 2C:3C] ----
    launch_gemm(stream, ws + OFF_FR, out1_w, ws + OFF_CAT3, nullptr, nullptr, nullptr, rgb,
                M, C, Din, Din, 0, 0, 0, 1, 3 * C * HW, HW, C, C * HW, 2);
    launch_gemm(stream, ws + OFF_FT, out2_w, ws + OFF_CAT3, nullptr, nullptr, nullptr, tin,
                M, C, Din, Din, 0, 0, 0, 1, 3 * C * HW, HW, 2 * C, C * HW, 2);

    // ---- c2 1x1 conv (+BN+ReLU) over cat3 -> final output (NCHW) ----
    launch_gemm(stream, ws + OFF_CAT3, c2_w, out + out_off, c2_b, bn2_g, bn2_b, nullptr,
                M, C, 3 * C, 0, 1, 3 * C * HW, HW, 1, C * HW, HW, 0, 0, 1);

    out_off += (long)M * C;
  }
}